// ScaledDotProductAttention_78512002171347
// MI455X (gfx1250) — compile-verified
//
#include <hip/hip_runtime.h>

// ScaledDotProductAttention for BS=4, H=16, S=1024, DK=64, f32 in/out.
// Outputs: context [BS,H,S,DK] then attn [BS,H,S,S], concatenated in d_out.
//
// Strategy (MI455X / gfx1250, wave32):
//  - 1 block = 4 waves = 128 threads handles one (b, h, 16-query-row) tile.
//  - QK^T and PV both run on V_WMMA_F32_16X16X4_F32 (f32 matrix pipe).
//  - Scores (16 x 1024 f32 = 64KB) live in LDS (dynamic, ~70KB of the 320KB WGP LDS)
//    so softmax reads them exactly once; normalized probs go straight to the
//    attn output and are reused from LDS for the PV GEMM.
//  - K input is pre-transposed [DK,S] so B-fragments of QK^T are coalesced loads.

#define BSZ   4
#define NH    16
#define SEQ   1024
#define DKD   64
#define QTILE 16
#define NWAVE 4
#define NTHR  128

typedef float v2f __attribute__((ext_vector_type(2)));
typedef float v8f __attribute__((ext_vector_type(8)));

// LDS layout in floats
#define SM_Q    0                    // 16*64   = 1024
#define SM_P    (SM_Q + QTILE*DKD)   // 16*1024 = 16384
#define SM_RED  (SM_P + QTILE*SEQ)   // 16*8    = 128
#define SM_RMAX (SM_RED + 16*8)      // 16
#define SM_RSUM (SM_RMAX + 16)       // 16
#define SM_TOT  (SM_RSUM + 16)       // 17568 floats = 70272 bytes

__global__ __launch_bounds__(NTHR)
void sdpa_wmma_f32_kernel(const float* __restrict__ q,
                          const float* __restrict__ k,     // [BS,H,DK,S]
                          const float* __restrict__ v,     // [BS,H,S,DK]
                          const float* __restrict__ prev,  // [BS,H,S,S]
                          const float* __restrict__ amask, // [S,S] additive causal
                          const unsigned char* __restrict__ kpm, // [BS,S] bool
                          float* __restrict__ out_ctx,     // [BS,H,S,DK]
                          float* __restrict__ out_attn) {  // [BS,H,S,S]
    extern __shared__ float sm[];
    float* Qs   = sm + SM_Q;
    float* Ps   = sm + SM_P;
    float* red  = sm + SM_RED;
    float* rmax = sm + SM_RMAX;
    float* rsum = sm + SM_RSUM;

    const int tid  = threadIdx.x;
    const int wave = tid >> 5;
    const int lane = tid & 31;
    const int l15  = lane & 15;
    const int hi   = lane >> 4;     // 0: lanes 0-15, 1: lanes 16-31
    const int ksel = hi << 1;       // A/B fragment K-offset (ISA 16x4 layout)

    const int blk = blockIdx.x;         // (b*H + h)*64 + qt
    const int qt  = blk & 63;           // SEQ/QTILE = 64
    const int bh  = blk >> 6;
    const int b   = bh >> 4;            // NH = 16
    const int q0  = qt * QTILE;

    const float* Qb = q    + ((size_t)bh * SEQ + q0) * DKD;
    const float* Kb = k    + (size_t)bh * DKD * SEQ;
    const float* Vb = v    + (size_t)bh * SEQ * DKD;
    const float* Pb = prev + ((size_t)bh * SEQ + q0) * SEQ;
    const float* Mb = amask + (size_t)q0 * SEQ;
    const unsigned char* KPb = kpm + (size_t)b * SEQ;

    // ---- load Q tile into LDS, pre-scaled by 1/sqrt(DK) = 0.125 ----
    for (int i = tid; i < QTILE * DKD; i += NTHR) Qs[i] = Qb[i] * 0.125f;
    __syncthreads();

    // ---- scores = QK^T/sqrt(d) + prev + attn_mask, padding -> -inf ----
    // Each wave owns 16 contiguous 16-wide key tiles.
    for (int t = 0; t < 16; ++t) {
        const int n0 = (wave * 16 + t) * 16;
        v8f c = {};
#pragma unroll
        for (int kk = 0; kk < DKD; kk += 4) {
            const int kb = kk + ksel;
            v2f a, bb;
            a.x  = Qs[l15 * DKD + kb];
            a.y  = Qs[l15 * DKD + kb + 1];
            bb.x = Kb[(size_t)kb * SEQ + n0 + l15];
            bb.y = Kb[(size_t)(kb + 1) * SEQ + n0 + l15];
            c = __builtin_amdgcn_wmma_f32_16x16x4_f32(false, a, false, bb,
                                                      (short)0, c, false, false);
        }
        const int col = n0 + l15;
        const bool padded = KPb[col] != 0;
#pragma unroll
        for (int r = 0; r < 8; ++r) {
            const int m = r + hi * 8;
            float s = c[r] + Pb[(size_t)m * SEQ + col] + Mb[(size_t)m * SEQ + col];
            if (padded) s = -__builtin_inff();
            Ps[m * SEQ + col] = s;
        }
    }
    __syncthreads();

    // ---- row softmax in LDS; write normalized probs to attn output ----
    {
        const int row = tid >> 3;   // 16 rows x 8 threads
        const int sub = tid & 7;
        float m = -__builtin_inff();
        for (int c = sub; c < SEQ; c += 8) m = fmaxf(m, Ps[row * SEQ + c]);
        red[row * 8 + sub] = m;
        __syncthreads();
        if (sub == 0) {
            float mm = red[row * 8];
#pragma unroll
            for (int i = 1; i < 8; ++i) mm = fmaxf(mm, red[row * 8 + i]);
            rmax[row] = mm;
        }
        __syncthreads();
        const float mm = rmax[row];
        float acc = 0.f;
        for (int c = sub; c < SEQ; c += 8) {
            float e = __expf(Ps[row * SEQ + c] - mm);
            Ps[row * SEQ + c] = e;
            acc += e;
        }
        red[row * 8 + sub] = acc;
        __syncthreads();
        if (sub == 0) {
            float s = 0.f;
#pragma unroll
            for (int i = 0; i < 8; ++i) s += red[row * 8 + i];
            rsum[row] = s;
        }
        __syncthreads();
        const float inv = 1.0f / rsum[row];
        float* aout = out_attn + ((size_t)bh * SEQ + q0 + row) * SEQ;
        for (int c = sub; c < SEQ; c += 8) {
            float p = Ps[row * SEQ + c] * inv;
            Ps[row * SEQ + c] = p;
            aout[c] = p;
        }
    }
    __syncthreads();

    // ---- context = P @ V ; each wave owns one 16-wide slice of DK ----
    {
        const int n0 = wave * 16;   // NWAVE*16 == DKD
        v8f c = {};
#pragma unroll 4
        for (int kk = 0; kk < SEQ; kk += 4) {
            const int kb = kk + ksel;
            v2f a, bb;
            a.x  = Ps[l15 * SEQ + kb];
            a.y  = Ps[l15 * SEQ + kb + 1];
            bb.x = Vb[(size_t)kb * DKD + n0 + l15];
            bb.y = Vb[(size_t)(kb + 1) * DKD + n0 + l15];
            c = __builtin_amdgcn_wmma_f32_16x16x4_f32(false, a, false, bb,
                                                      (short)0, c, false, false);
        }
#pragma unroll
        for (int r = 0; r < 8; ++r) {
            const int m = r + hi * 8;
            out_ctx[((size_t)bh * SEQ + q0 + m) * DKD + n0 + l15] = c[r];
        }
    }
}

extern "C" void kernel_launch(void* const* d_in, const int* in_sizes, int n_in,
                              void* d_out, int out_size, void* d_ws, size_t ws_size,
                              hipStream_t stream) {
    const float* q     = (const float*)d_in[0];
    const float* k     = (const float*)d_in[1];
    const float* v     = (const float*)d_in[2];
    const float* prev  = (const float*)d_in[3];
    const float* amask = (const float*)d_in[4];
    const unsigned char* kpm = (const unsigned char*)d_in[5]; // JAX bool = 1 byte

    float* out_ctx  = (float*)d_out;
    float* out_attn = out_ctx + (size_t)BSZ * NH * SEQ * DKD;

    const dim3 grid(BSZ * NH * (SEQ / QTILE));
    const dim3 block(NTHR);
    const size_t smem = (size_t)SM_TOT * sizeof(float);

    sdpa_wmma_f32_kernel<<<grid, block, smem, stream>>>(q, k, v, prev, amask, kpm,
                                                        out_ctx, out_attn);
}